// FAM_GCN_58626303591152
// MI455X (gfx1250) — compile-verified
//
#include <hip/hip_runtime.h>
#include <hip/hip_bf16.h>

typedef __attribute__((ext_vector_type(2))) float v2f;
typedef __attribute__((ext_vector_type(8))) float v8f;

#define N_NODES 100000   // multiple of 32: GEMM grid covers rows exactly
#define N_EDGES 1250000
#define IN_DIM  32
#define H_DIM   64
#define OUT_DIM 64

static_assert(N_NODES % 32 == 0, "GEMM kernels assume N is a multiple of 32");

// ---------------------------------------------------------------------------
// 1) structural degrees via float atomics (L2-resident accumulators)
// ---------------------------------------------------------------------------
__global__ void fam_deg_kernel(const int* __restrict__ src, const int* __restrict__ dst,
                               float* __restrict__ outdeg, float* __restrict__ indeg, int E) {
    int e = blockIdx.x * blockDim.x + threadIdx.x;
    if (e < E) {
        atomicAdd(&outdeg[src[e]], 1.0f);
        atomicAdd(&indeg[dst[e]], 1.0f);
    }
}

// deg -> clamped symmetric norm, in place
__global__ void fam_norm_kernel(float* __restrict__ ns, float* __restrict__ nd, int N) {
    int i = blockIdx.x * blockDim.x + threadIdx.x;
    if (i < N) {
        ns[i] = rsqrtf(fmaxf(ns[i], 1.0f));
        nd[i] = rsqrtf(fmaxf(nd[i], 1.0f));
    }
}

// ---------------------------------------------------------------------------
// 2) TS fuzzy attention: att = exp(-(||f_s - f_d|| - mu_t)^2 / (2 sigma_t^2))
// ---------------------------------------------------------------------------
__global__ void fam_att_kernel(const float* __restrict__ feat,
                               const int* __restrict__ src, const int* __restrict__ dst,
                               const int* __restrict__ etypes,
                               const float* __restrict__ mu, const float* __restrict__ sigma,
                               float* __restrict__ att, int E) {
    int e = blockIdx.x * blockDim.x + threadIdx.x;
    if (e >= E) return;
    const float* fs = feat + (long)src[e] * IN_DIM;
    const float* fd = feat + (long)dst[e] * IN_DIM;
    float acc = 0.0f;
#pragma unroll
    for (int i = 0; i < IN_DIM; i += 4) {
        float4 a = *(const float4*)(fs + i);
        float4 b = *(const float4*)(fd + i);
        float dx = a.x - b.x, dy = a.y - b.y, dz = a.z - b.z, dw = a.w - b.w;
        acc += dx * dx + dy * dy + dz * dz + dw * dw;
    }
    float dist = sqrtf(acc);
    int t = etypes[e];
    float z = dist - mu[t];
    float sg = sigma[t];
    att[e] = expf(-(z * z) / (2.0f * sg * sg));
}

// ---------------------------------------------------------------------------
// 3) edge scatter: agg[dst] += h[src] * norm_src[src] * att[e]
//    one lane per (edge, dim): coalesced row gather, coalesced f32 atomics
// ---------------------------------------------------------------------------
template <int LOGD>
__global__ void fam_scatter_kernel(const float* __restrict__ h,
                                   const int* __restrict__ src, const int* __restrict__ dst,
                                   const float* __restrict__ norm_src,
                                   const float* __restrict__ att,
                                   float* __restrict__ agg, int E) {
    const int D = 1 << LOGD;
    long gid = (long)blockIdx.x * blockDim.x + threadIdx.x;
    long e = gid >> LOGD;
    int d = (int)(gid & (D - 1));
    if (e >= E) return;
    int s = src[e];
    int t = dst[e];
    float v = h[(long)s * D + d] * norm_src[s] * att[e];
    atomicAdd(&agg[(long)t * D + d], v);
}

// ---------------------------------------------------------------------------
// 4) fused (agg * norm_dst) @ W + b, ReLU — f32 WMMA 16x16x4 per wave tile.
//    Block = 256 threads = 8 waves; block owns 32 rows; wave (rt,ct) owns a
//    16x16 tile of the [32 x 64] output. X tile and W staged in LDS.
//    N is an exact multiple of 32, so no bounds checks -> branchless epilogue.
// ---------------------------------------------------------------------------
template <int K>
__global__ __launch_bounds__(256)
void fam_gemm_bias_relu_kernel(const float* __restrict__ agg,
                               const float* __restrict__ norm_dst,
                               const float* __restrict__ W,
                               const float* __restrict__ bias,
                               float* __restrict__ out) {
    __shared__ float sX[32 * K];
    __shared__ float sW[K * 64];
    const int tid = threadIdx.x;
    const int row0 = blockIdx.x * 32;

    for (int i = tid; i < K * 64; i += 256) sW[i] = W[i];
    for (int i = tid; i < 32 * K; i += 256) {
        int r = i / K, c = i % K;
        int row = row0 + r;
        sX[i] = agg[(long)row * K + c] * norm_dst[row];
    }
    __syncthreads();

    const int wave = tid >> 5;
    const int lane = tid & 31;
    const int rt = wave >> 2;        // row tile 0..1
    const int ct = wave & 3;         // col tile 0..3
    const int m = lane & 15;
    const int kh = (lane >> 4) << 1; // 0 (lanes 0-15) or 2 (lanes 16-31)

    v8f acc = {};
#pragma unroll
    for (int ks = 0; ks < K; ks += 4) {
        v2f a, b;
        // A 16x4 layout: VGPR0 = K (kh), VGPR1 = K (kh+1); M = lane&15
        a.x = sX[(rt * 16 + m) * K + ks + kh];
        a.y = sX[(rt * 16 + m) * K + ks + kh + 1];
        // B 4x16 layout: VGPR0 = row (kh), VGPR1 = row (kh+1); N = lane&15
        b.x = sW[(ks + kh) * 64 + ct * 16 + m];
        b.y = sW[(ks + kh + 1) * 64 + ct * 16 + m];
        acc = __builtin_amdgcn_wmma_f32_16x16x4_f32(
            false, a, false, b, (short)0, acc, false, false);
    }

    const float bb = bias[ct * 16 + m];
    float* outp = out + (long)(row0 + rt * 16 + ((lane >> 4) << 3)) * 64 + ct * 16 + m;
#pragma unroll
    for (int i = 0; i < 8; ++i) {
        // C/D layout: VGPR i -> M = i + 8*(lane>=16), N = lane&15
        float v = acc[i] + bb;
        outp[(long)i * 64] = v > 0.0f ? v : 0.0f;
    }
}

// ---------------------------------------------------------------------------
// 5) output reduction over rows 2..N-1 (post-ReLU => values >= 0, so float
//    max/min order == int order on the bit patterns)
// ---------------------------------------------------------------------------
__global__ void fam_initred_kernel(float* __restrict__ rsum,
                                   unsigned* __restrict__ rmax,
                                   unsigned* __restrict__ rmin) {
    int d = threadIdx.x;
    if (d < OUT_DIM) {
        rsum[d] = 0.0f;
        rmax[d] = 0u;           // 0.0f
        rmin[d] = 0x7f800000u;  // +inf
    }
}

__global__ void fam_reduce_kernel(const float* __restrict__ x2, int N,
                                  float* __restrict__ rsum,
                                  unsigned* __restrict__ rmax,
                                  unsigned* __restrict__ rmin) {
    __shared__ float ssum[256], smax[256], smin[256];
    const int tid = threadIdx.x;
    const int d = tid & 63;
    const int sub = tid >> 6;  // 0..3
    const int row0 = 2 + blockIdx.x * 1024;
    const int rowEnd = (row0 + 1024 < N) ? row0 + 1024 : N;
    float lsum = 0.0f, lmax = 0.0f, lmin = __int_as_float(0x7f800000);
    for (int r = row0 + sub; r < rowEnd; r += 4) {
        float v = x2[(long)r * 64 + d];
        lsum += v;
        lmax = fmaxf(lmax, v);
        lmin = fminf(lmin, v);
    }
    ssum[tid] = lsum; smax[tid] = lmax; smin[tid] = lmin;
    __syncthreads();
    if (tid < 64) {
#pragma unroll
        for (int i = 1; i < 4; ++i) {
            lsum += ssum[tid + 64 * i];
            lmax = fmaxf(lmax, smax[tid + 64 * i]);
            lmin = fminf(lmin, smin[tid + 64 * i]);
        }
        atomicAdd(&rsum[d], lsum);
        atomicMax((int*)&rmax[d], __float_as_int(lmax));
        atomicMin((int*)&rmin[d], __float_as_int(lmin));
    }
}

__global__ void fam_final_kernel(const float* __restrict__ x2,
                                 const float* __restrict__ rsum,
                                 const unsigned* __restrict__ rmax,
                                 const unsigned* __restrict__ rmin,
                                 float* __restrict__ out, int N) {
    int d = threadIdx.x;
    if (d < OUT_DIM) {
        out[0 * OUT_DIM + d] = x2[0 * 64 + d];
        out[1 * OUT_DIM + d] = x2[1 * 64 + d];
        out[2 * OUT_DIM + d] = rsum[d] / (float)(N - 2);
        out[3 * OUT_DIM + d] = __int_as_float((int)rmax[d]);
        out[4 * OUT_DIM + d] = __int_as_float((int)rmin[d]);
    }
}

// ---------------------------------------------------------------------------
extern "C" void kernel_launch(void* const* d_in, const int* in_sizes, int n_in,
                              void* d_out, int out_size, void* d_ws, size_t ws_size,
                              hipStream_t stream) {
    const float* feat   = (const float*)d_in[0];
    const int*   src    = (const int*)d_in[1];
    const int*   dst    = (const int*)d_in[2];
    const int*   etypes = (const int*)d_in[3];
    const float* W1     = (const float*)d_in[4];
    const float* b1     = (const float*)d_in[5];
    const float* W2     = (const float*)d_in[6];
    const float* b2     = (const float*)d_in[7];
    const float* mu     = (const float*)d_in[8];
    const float* sigma  = (const float*)d_in[9];
    float* out = (float*)d_out;

    const int N = N_NODES;
    const int E = N_EDGES;

    // carve workspace (256B-aligned regions)
    char* p = (char*)d_ws;
    auto carve = [&](size_t bytes) -> void* {
        void* r = (void*)p;
        p += (bytes + 255) & ~(size_t)255;
        return r;
    };
    float*    att      = (float*)carve((size_t)E * 4);
    float*    norm_src = (float*)carve((size_t)N * 4);  // holds out-degree first
    float*    norm_dst = (float*)carve((size_t)N * 4);  // holds in-degree first
    float*    agg      = (float*)carve((size_t)N * 64 * 4);
    float*    x1       = (float*)carve((size_t)N * 64 * 4);
    float*    x2       = (float*)carve((size_t)N * 64 * 4);
    float*    rsum     = (float*)carve(64 * 4);
    unsigned* rmax     = (unsigned*)carve(64 * 4);
    unsigned* rmin     = (unsigned*)carve(64 * 4);

    const int T = 256;

    // degrees -> norms
    hipMemsetAsync(norm_src, 0, (size_t)N * 4, stream);
    hipMemsetAsync(norm_dst, 0, (size_t)N * 4, stream);
    fam_deg_kernel<<<(E + T - 1) / T, T, 0, stream>>>(src, dst, norm_src, norm_dst, E);
    fam_norm_kernel<<<(N + T - 1) / T, T, 0, stream>>>(norm_src, norm_dst, N);

    // fuzzy attention
    fam_att_kernel<<<(E + T - 1) / T, T, 0, stream>>>(feat, src, dst, etypes, mu, sigma, att, E);

    // layer 1: scatter (D=32) then WMMA GEMM 32->64 with bias+ReLU
    hipMemsetAsync(agg, 0, (size_t)N * IN_DIM * 4, stream);
    {
        long work = (long)E * IN_DIM;
        fam_scatter_kernel<5><<<(int)((work + T - 1) / T), T, 0, stream>>>(
            feat, src, dst, norm_src, att, agg, E);
    }
    fam_gemm_bias_relu_kernel<IN_DIM><<<N / 32, T, 0, stream>>>(
        agg, norm_dst, W1, b1, x1);

    // layer 2: scatter (D=64) then WMMA GEMM 64->64 with bias+ReLU
    hipMemsetAsync(agg, 0, (size_t)N * H_DIM * 4, stream);
    {
        long work = (long)E * H_DIM;
        fam_scatter_kernel<6><<<(int)((work + T - 1) / T), T, 0, stream>>>(
            x1, src, dst, norm_src, att, agg, E);
    }
    fam_gemm_bias_relu_kernel<H_DIM><<<N / 32, T, 0, stream>>>(
        agg, norm_dst, W2, b2, x2);

    // output reduction: rows 2..N-1 -> mean/max/min, plus rows 0,1 verbatim
    fam_initred_kernel<<<1, 64, 0, stream>>>(rsum, rmax, rmin);
    int nred = (N - 2 + 1023) / 1024;
    fam_reduce_kernel<<<nred, T, 0, stream>>>(x2, N, rsum, rmax, rmin);
    fam_final_kernel<<<1, 64, 0, stream>>>(x2, rsum, rmax, rmin, out, N);

    (void)in_sizes; (void)n_in; (void)out_size; (void)ws_size;
}